// CPNet_25821343384069
// MI455X (gfx1250) — compile-verified
//
#include <hip/hip_runtime.h>
#include <math.h>

#define EPSF 1e-5f
#define FACTF 2.0f

constexpr int B_ = 8;
constexpr int N_ = 4096;
constexpr int THREADS = 256;                  // 8 waves (wave32)
constexpr int ROWS_PER_BLOCK = 128;           // 8 waves * 16 rows
constexpr int CHUNKS = N_ / ROWS_PER_BLOCK;   // 32

typedef float v2f __attribute__((ext_vector_type(2)));
typedef float v8f __attribute__((ext_vector_type(8)));

// ---------------------------------------------------------------------------
// Kernel 1: soft nearest-neighbor (attention-like). One wave owns a 16-row
// tile of pc1; loops over 256 column tiles of pc2 staged in LDS.
// d-tile computed with V_WMMA_F32_16X16X4_F32:
//   d = [x1,y1,z1,1]·[-2x2;-2y2;-2z2;|p2|^2]  +  [|p1|^2,0,0,0]·[1;0;0;0]
// Pass 1: row-min of d (for softmax max-score). Pass 2: weighted sums.
// ---------------------------------------------------------------------------
__global__ __launch_bounds__(THREADS)
void cpnet_soft_nn(const float* __restrict__ pc1,
                   const float* __restrict__ pc2,
                   float* __restrict__ nearest /* [3][B][N] */) {
    __shared__ float s_x2[N_];
    __shared__ float s_y2[N_];
    __shared__ float s_z2[N_];

    const int b     = blockIdx.x / CHUNKS;
    const int chunk = blockIdx.x % CHUNKS;

    const float* p2b = pc2 + (size_t)b * 4 * N_;
    for (int n = threadIdx.x; n < N_; n += THREADS) {
        s_x2[n] = p2b[0 * N_ + n];
        s_y2[n] = p2b[1 * N_ + n];
        s_z2[n] = p2b[2 * N_ + n];
    }
    __syncthreads();

    const int  wave   = threadIdx.x >> 5;
    const int  lane   = threadIdx.x & 31;
    const int  laneLo = lane & 15;
    const bool hiHalf = lane >= 16;
    const int  rowBase = chunk * ROWS_PER_BLOCK + wave * 16;

    // A-matrix (16x4 f32): lanes 0-15 hold {K0,K1}, lanes 16-31 hold {K2,K3},
    // M = lane%16. Fixed for the whole column loop.
    const float* p1b = pc1 + (size_t)b * 4 * N_;
    const int   r   = rowBase + laneLo;
    const float x1  = p1b[0 * N_ + r];
    const float y1  = p1b[1 * N_ + r];
    const float z1  = p1b[2 * N_ + r];
    const float r1s = x1 * x1 + y1 * y1 + z1 * z1;

    v2f Am  = { hiHalf ? z1 : x1,    hiHalf ? 1.0f : y1 };   // [x1,y1,z1,1]
    v2f Am2 = { hiHalf ? 0.0f : r1s, 0.0f };                 // [|p1|^2,0,0,0]
    v2f Bm2 = { hiHalf ? 0.0f : 1.0f, 0.0f };                // [1;0;0;0]

    // ---------------- Pass 1: per-row min of d ----------------
    float dmin[8];
#pragma unroll
    for (int i = 0; i < 8; ++i) dmin[i] = 3.4e38f;

    for (int mt = 0; mt < N_ / 16; ++mt) {
        const int   col = (mt << 4) + laneLo;
        const float bx = s_x2[col], by = s_y2[col], bz = s_z2[col];
        const float br = bx * bx + by * by + bz * bz;
        // B-matrix (4x16): lanes=N, VGPR0={K0|K2}, VGPR1={K1|K3}
        v2f Bm = { hiHalf ? -2.0f * bz : -2.0f * bx,
                   hiHalf ? br         : -2.0f * by };
        v8f c = {0.f, 0.f, 0.f, 0.f, 0.f, 0.f, 0.f, 0.f};
        c = __builtin_amdgcn_wmma_f32_16x16x4_f32(false, Am,  false, Bm,  (short)0, c, false, false);
        c = __builtin_amdgcn_wmma_f32_16x16x4_f32(false, Am2, false, Bm2, (short)0, c, false, false);
#pragma unroll
        for (int i = 0; i < 8; ++i) dmin[i] = fminf(dmin[i], c[i]);
    }
    // reduce min over the 16 lanes that share each row (butterfly within half)
#pragma unroll
    for (int i = 0; i < 8; ++i) {
#pragma unroll
        for (int off = 1; off < 16; off <<= 1)
            dmin[i] = fminf(dmin[i], __shfl_xor(dmin[i], off, 32));
    }
    float smax[8];
#pragma unroll
    for (int i = 0; i < 8; ++i)
        smax[i] = __fdividef(FACTF, fmaxf(dmin[i], EPSF));

    // ---------------- Pass 2: softmax-weighted sums ----------------
    float wsum[8], ax[8], ay[8], az[8];
#pragma unroll
    for (int i = 0; i < 8; ++i) { wsum[i] = 0.f; ax[i] = 0.f; ay[i] = 0.f; az[i] = 0.f; }

    for (int mt = 0; mt < N_ / 16; ++mt) {
        const int   col = (mt << 4) + laneLo;
        const float bx = s_x2[col], by = s_y2[col], bz = s_z2[col];
        const float br = bx * bx + by * by + bz * bz;
        v2f Bm = { hiHalf ? -2.0f * bz : -2.0f * bx,
                   hiHalf ? br         : -2.0f * by };
        v8f c = {0.f, 0.f, 0.f, 0.f, 0.f, 0.f, 0.f, 0.f};
        c = __builtin_amdgcn_wmma_f32_16x16x4_f32(false, Am,  false, Bm,  (short)0, c, false, false);
        c = __builtin_amdgcn_wmma_f32_16x16x4_f32(false, Am2, false, Bm2, (short)0, c, false, false);
#pragma unroll
        for (int i = 0; i < 8; ++i) {
            const float d = fmaxf(c[i], EPSF);
            const float p = __expf(__fdividef(FACTF, d) - smax[i]);
            wsum[i] += p;
            ax[i] += p * bx;
            ay[i] += p * by;
            az[i] += p * bz;
        }
    }
    // reduce sums over the 16 lanes that share each row
#pragma unroll
    for (int i = 0; i < 8; ++i) {
#pragma unroll
        for (int off = 1; off < 16; off <<= 1) {
            wsum[i] += __shfl_xor(wsum[i], off, 32);
            ax[i]   += __shfl_xor(ax[i],   off, 32);
            ay[i]   += __shfl_xor(ay[i],   off, 32);
            az[i]   += __shfl_xor(az[i],   off, 32);
        }
    }
    // row of vgpr i: rowBase + i (lanes 0-15) or rowBase + i + 8 (lanes 16-31)
    float* nx = nearest + (size_t)b * N_;
    float* ny = nearest + (size_t)B_ * N_ + (size_t)b * N_;
    float* nz = nearest + 2 * (size_t)B_ * N_ + (size_t)b * N_;
#pragma unroll
    for (int i = 0; i < 8; ++i) {
        if (laneLo == i) {
            const int   rr  = rowBase + i + (hiHalf ? 8 : 0);
            const float inv = __fdividef(1.0f, wsum[i]);
            nx[rr] = ax[i] * inv;
            ny[rr] = ay[i] * inv;
            nz[rr] = az[i] * inv;
        }
    }
}

// ---------------------------------------------------------------------------
// Kernel 2: per-batch epilogue. One block per batch.
// ---------------------------------------------------------------------------
__device__ __forceinline__ float det3(const float M[3][3]) {
    return M[0][0] * (M[1][1] * M[2][2] - M[1][2] * M[2][1])
         - M[0][1] * (M[1][0] * M[2][2] - M[1][2] * M[2][0])
         + M[0][2] * (M[1][0] * M[2][1] - M[1][1] * M[2][0]);
}

__global__ __launch_bounds__(THREADS)
void cpnet_finalize(const float* __restrict__ pc1,
                    const float* __restrict__ nearest,
                    float* __restrict__ out) {
    const int b   = blockIdx.x;
    const int tid = threadIdx.x;

    __shared__ float s_dist[N_];
    __shared__ float s_red[THREADS / 32];
    __shared__ float s_mean;
    __shared__ float s_acc[16];

    const float* p1b = pc1 + (size_t)b * 4 * N_;
    const float* nx  = nearest + (size_t)b * N_;
    const float* ny  = nearest + (size_t)B_ * N_ + (size_t)b * N_;
    const float* nz  = nearest + 2 * (size_t)B_ * N_ + (size_t)b * N_;

    // pass 1: distances + mean
    float lsum = 0.f;
    for (int n = tid; n < N_; n += THREADS) {
        const float dx = p1b[n] - nx[n];
        const float dy = p1b[N_ + n] - ny[n];
        const float dz = p1b[2 * N_ + n] - nz[n];
        const float d  = sqrtf(dx * dx + dy * dy + dz * dz);
        s_dist[n] = d;
        lsum += d;
    }
#pragma unroll
    for (int off = 16; off >= 1; off >>= 1) lsum += __shfl_xor(lsum, off, 32);
    if ((tid & 31) == 0) s_red[tid >> 5] = lsum;
    if (tid < 16) s_acc[tid] = 0.f;
    __syncthreads();
    if (tid == 0) {
        float t = 0.f;
        for (int w = 0; w < THREADS / 32; ++w) t += s_red[w];
        s_mean = t / (float)N_;
    }
    __syncthreads();
    const float mean = s_mean;

    // pass 2: inlier mask + raw moments (wsum, S1, S2, M=sum idx*p1*p2^T)
    float acc[16];
#pragma unroll
    for (int k = 0; k < 16; ++k) acc[k] = 0.f;
    for (int n = tid; n < N_; n += THREADS) {
        const float dist = s_dist[n];
        // sigmoid((dist - mean - EPS) * -1e10)
        const float z   = (dist - mean * 1.0f - EPSF) * 1e10f;
        const float idx = __fdividef(1.0f, 1.0f + __expf(z));
        const float X1 = p1b[n], Y1 = p1b[N_ + n], Z1 = p1b[2 * N_ + n];
        const float X2 = nx[n],  Y2 = ny[n],       Z2 = nz[n];
        acc[0]  += idx;
        acc[1]  += idx * X1;  acc[2]  += idx * Y1;  acc[3]  += idx * Z1;
        acc[4]  += idx * X2;  acc[5]  += idx * Y2;  acc[6]  += idx * Z2;
        acc[7]  += idx * X1 * X2;  acc[8]  += idx * X1 * Y2;  acc[9]  += idx * X1 * Z2;
        acc[10] += idx * Y1 * X2;  acc[11] += idx * Y1 * Y2;  acc[12] += idx * Y1 * Z2;
        acc[13] += idx * Z1 * X2;  acc[14] += idx * Z1 * Y2;  acc[15] += idx * Z1 * Z2;
    }
#pragma unroll
    for (int k = 0; k < 16; ++k) {
#pragma unroll
        for (int off = 16; off >= 1; off >>= 1) acc[k] += __shfl_xor(acc[k], off, 32);
        if ((tid & 31) == 0) atomicAdd(&s_acc[k], acc[k]);
    }
    __syncthreads();

    if (tid != 0) return;

    // ---- solve Kabsch on 3x3 H (thread 0 only) ----
    const float wsum = s_acc[0];
    const float invw = __fdividef(1.0f, wsum);
    float c1[3] = { s_acc[1] * invw, s_acc[2] * invw, s_acc[3] * invw };
    float c2[3] = { s_acc[4] * invw, s_acc[5] * invw, s_acc[6] * invw };
    float H[3][3];
    for (int a = 0; a < 3; ++a)
        for (int d = 0; d < 3; ++d)
            H[a][d] = s_acc[7 + a * 3 + d] - wsum * c1[a] * c2[d];

    // Jacobi eigen of S = H^T H  ->  V, eigenvalues
    float S[3][3], V[3][3];
    for (int a = 0; a < 3; ++a)
        for (int d = 0; d < 3; ++d) {
            float s = 0.f;
            for (int k = 0; k < 3; ++k) s += H[k][a] * H[k][d];
            S[a][d] = s;
            V[a][d] = (a == d) ? 1.f : 0.f;
        }
    for (int sweep = 0; sweep < 15; ++sweep) {
        const int PQ[3][2] = { {0, 1}, {0, 2}, {1, 2} };
        for (int e = 0; e < 3; ++e) {
            const int p = PQ[e][0], q = PQ[e][1];
            const float apq = S[p][q];
            if (fabsf(apq) < 1e-25f) continue;
            const float tau = (S[q][q] - S[p][p]) / (2.0f * apq);
            const float t   = (tau >= 0.f ? 1.f : -1.f) /
                              (fabsf(tau) + sqrtf(1.f + tau * tau));
            const float cc  = rsqrtf(1.f + t * t);
            const float ss  = t * cc;
            for (int k = 0; k < 3; ++k) {
                const float skp = S[k][p], skq = S[k][q];
                S[k][p] = cc * skp - ss * skq;
                S[k][q] = ss * skp + cc * skq;
            }
            for (int k = 0; k < 3; ++k) {
                const float spk = S[p][k], sqk = S[q][k];
                S[p][k] = cc * spk - ss * sqk;
                S[q][k] = ss * spk + cc * sqk;
            }
            for (int k = 0; k < 3; ++k) {
                const float vkp = V[k][p], vkq = V[k][q];
                V[k][p] = cc * vkp - ss * vkq;
                V[k][q] = ss * vkp + cc * vkq;
            }
        }
    }
    // sort eigenvalues descending, permute V columns
    float ev[3] = { S[0][0], S[1][1], S[2][2] };
    for (int a = 0; a < 2; ++a)
        for (int d = a + 1; d < 3; ++d)
            if (ev[d] > ev[a]) {
                float tmp = ev[a]; ev[a] = ev[d]; ev[d] = tmp;
                for (int k = 0; k < 3; ++k) {
                    float tv = V[k][a]; V[k][a] = V[k][d]; V[k][d] = tv;
                }
            }
    // U columns: u_i = H v_i / |H v_i|  (fallback cross for the smallest)
    float U[3][3];
    float n0 = 0.f;
    for (int i = 0; i < 3; ++i) {
        float u[3];
        for (int a = 0; a < 3; ++a)
            u[a] = H[a][0] * V[0][i] + H[a][1] * V[1][i] + H[a][2] * V[2][i];
        float nn = sqrtf(u[0] * u[0] + u[1] * u[1] + u[2] * u[2]);
        if (i == 0) n0 = nn;
        if (i == 2 && nn <= 1e-7f * n0) {
            u[0] = U[1][0] * U[2][1] - U[2][0] * U[1][1];
            u[1] = U[2][0] * U[0][1] - U[0][0] * U[2][1];
            u[2] = U[0][0] * U[1][1] - U[1][0] * U[0][1];
            nn = sqrtf(u[0] * u[0] + u[1] * u[1] + u[2] * u[2]);
        }
        const float inv = __fdividef(1.0f, fmaxf(nn, 1e-30f));
        for (int a = 0; a < 3; ++a) U[a][i] = u[a] * inv;
    }
    const float sign = (det3(U) * det3(V) < 0.f) ? -1.f : 1.f;
    for (int a = 0; a < 3; ++a) V[a][2] *= sign;

    float R[3][3];
    for (int a = 0; a < 3; ++a)
        for (int d = 0; d < 3; ++d)
            R[a][d] = V[a][0] * U[d][0] + V[a][1] * U[d][1] + V[a][2] * U[d][2];

    float tvec[3];
    for (int a = 0; a < 3; ++a)
        tvec[a] = c2[a] - (R[a][0] * c1[0] + R[a][1] * c1[1] + R[a][2] * c1[2]);

    // outputs: T[B][4][4] | q[B][4] | t[B][3]
    float* T = out + (size_t)b * 16;
    for (int a = 0; a < 3; ++a) {
        for (int d = 0; d < 3; ++d) T[a * 4 + d] = R[a][d];
        T[a * 4 + 3] = tvec[a];
    }
    T[12] = 0.f; T[13] = 0.f; T[14] = 0.f; T[15] = 1.f;

    const float qw = 0.5f * sqrtf(fmaxf(1.f + R[0][0] + R[1][1] + R[2][2], 1e-12f));
    const float qx = 0.5f * sqrtf(fmaxf(1.f + R[0][0] - R[1][1] - R[2][2], 1e-12f)) *
                     ((R[2][1] - R[1][2] >= 0.f) ? 1.f : -1.f);
    const float qy = 0.5f * sqrtf(fmaxf(1.f - R[0][0] + R[1][1] - R[2][2], 1e-12f)) *
                     ((R[0][2] - R[2][0] >= 0.f) ? 1.f : -1.f);
    const float qz = 0.5f * sqrtf(fmaxf(1.f - R[0][0] - R[1][1] + R[2][2], 1e-12f)) *
                     ((R[1][0] - R[0][1] >= 0.f) ? 1.f : -1.f);
    float* q = out + (size_t)B_ * 16 + (size_t)b * 4;
    q[0] = qw; q[1] = qx; q[2] = qy; q[3] = qz;
    float* to = out + (size_t)B_ * 16 + (size_t)B_ * 4 + (size_t)b * 3;
    to[0] = tvec[0]; to[1] = tvec[1]; to[2] = tvec[2];
}

// ---------------------------------------------------------------------------
extern "C" void kernel_launch(void* const* d_in, const int* in_sizes, int n_in,
                              void* d_out, int out_size, void* d_ws, size_t ws_size,
                              hipStream_t stream) {
    (void)in_sizes; (void)n_in; (void)out_size; (void)ws_size;
    const float* pc1 = (const float*)d_in[0];
    const float* pc2 = (const float*)d_in[1];
    float* out     = (float*)d_out;
    float* nearest = (float*)d_ws;   // 3 * B * N floats = 384 KB

    cpnet_soft_nn<<<dim3(B_ * CHUNKS), dim3(THREADS), 0, stream>>>(pc1, pc2, nearest);
    cpnet_finalize<<<dim3(B_), dim3(THREADS), 0, stream>>>(pc1, nearest, out);
}